// EmbeddingShard_46488726012680
// MI455X (gfx1250) — compile-verified
//
#include <hip/hip_runtime.h>
#include <stdint.h>

typedef float v4f __attribute__((ext_vector_type(4)));

#define N_VOCAB 50400
#define D_MODEL 4096
#define SHARDS 8
#define SEQ 2048
#define BATCH 8
#define D_PER_SHARD 512

// ---------------------------------------------------------------------------
// Kernel 1: bias_sum[d] = sum over shards of b[s][d]   (4096 outputs, trivial)
// ---------------------------------------------------------------------------
__global__ void bias_sum_kernel(const float* __restrict__ b,
                                float* __restrict__ bias_sum) {
    int d = blockIdx.x * blockDim.x + threadIdx.x;   // grid sized exactly
    float acc = 0.f;
#pragma unroll
    for (int s = 0; s < SHARDS; ++s) acc += b[s * D_MODEL + d];
    bias_sum[d] = acc;
}

// ---------------------------------------------------------------------------
// Kernel 2: one block per seq position s.
//   - async-copy the interleaved pos row (16KB) into LDS (CDNA5 async path)
//   - prefetch the 8 gathered w rows (global_prefetch_b8)
//   - batch loop: out[b,s,:] = w[x[b,s],:] + (pos+bias)  via B128 + NT stores
// ---------------------------------------------------------------------------
__global__ void __launch_bounds__(512) embed_add_kernel(
    const int*   __restrict__ x,         // [BATCH, SEQ]
    const float* __restrict__ w,         // [N_VOCAB, D_MODEL]
    const float* __restrict__ pos,       // [SHARDS, SEQ, D_PER_SHARD]
    const float* __restrict__ bias_sum,  // [D_MODEL] (workspace, L2-hot)
    float*       __restrict__ out)       // [BATCH, SEQ, D_MODEL]
{
    __shared__ v4f lds4[D_MODEL / 4];    // 16 KB: pos row in output-d order
    const int s = blockIdx.x;
    const int t = threadIdx.x;

    // Stage pos row for this seq position into LDS via async global->LDS B128.
    // Output d index = shard*512 + j  ->  float4 slot idx = shard*128 + j/4,
    // so lds4[idx] holds d = idx*4..idx*4+3. 1024 float4s, 2 per lane.
    const uint64_t pos_base = (uint64_t)(uintptr_t)pos;
#pragma unroll
    for (int r = 0; r < 2; ++r) {
        const int idx    = t + r * 512;
        const int shard  = idx >> 7;
        const int within = idx & 127;
        const uint32_t goff =
            (uint32_t)((((shard * SEQ + s) * D_PER_SHARD) + within * 4) *
                       (int)sizeof(float));
        // low 32 bits of the generic address of a __shared__ object are the
        // LDS byte offset the async-load VDST operand expects
        const uint32_t lds_addr = (uint32_t)(uintptr_t)(&lds4[idx]);
        asm volatile("global_load_async_to_lds_b128 %0, %1, %2"
                     :
                     : "v"(lds_addr), "v"(goff), "s"(pos_base)
                     : "memory");
    }

    // Token ids are uniform per block -> scalar loads; then prefetch all 8
    // gathered rows (each lane covers 32B stride -> whole 16KB row).
    int toks[BATCH];
#pragma unroll
    for (int bb = 0; bb < BATCH; ++bb) toks[bb] = x[bb * SEQ + s];
#pragma unroll
    for (int bb = 0; bb < BATCH; ++bb)
        __builtin_prefetch(w + (uint64_t)toks[bb] * D_MODEL + (t << 3), 0, 3);

    // Loop-invariant (pos + bias) for this thread's two float4 slots.
    const int p0 = t, p1 = t + 512;
    v4f pb0 = *(const v4f*)(bias_sum + p0 * 4);
    v4f pb1 = *(const v4f*)(bias_sum + p1 * 4);

#if __has_builtin(__builtin_amdgcn_s_wait_asynccnt)
    __builtin_amdgcn_s_wait_asynccnt(0);
#else
    asm volatile("s_wait_asynccnt 0" ::: "memory");
#endif
    // Each thread reads exactly the LDS slots its own lane async-wrote, so a
    // per-wave asynccnt wait is sufficient (no workgroup barrier needed).
    pb0 += lds4[p0];
    pb1 += lds4[p1];

#pragma unroll
    for (int bb = 0; bb < BATCH; ++bb) {
        const float* wrow = w + (uint64_t)toks[bb] * D_MODEL;
        v4f w0 = *(const v4f*)(wrow + p0 * 4);
        v4f w1 = *(const v4f*)(wrow + p1 * 4);
        float* orow = out + ((uint64_t)bb * SEQ + s) * D_MODEL;
        // write-once 256MB stream: non-temporal so w/pos reuse stays in L2
        __builtin_nontemporal_store(w0 + pb0, (v4f*)(orow + p0 * 4));
        __builtin_nontemporal_store(w1 + pb1, (v4f*)(orow + p1 * 4));
    }
}

// ---------------------------------------------------------------------------
extern "C" void kernel_launch(void* const* d_in, const int* in_sizes, int n_in,
                              void* d_out, int out_size, void* d_ws,
                              size_t ws_size, hipStream_t stream) {
    (void)in_sizes; (void)n_in; (void)out_size; (void)ws_size;
    const int*   x   = (const int*)  d_in[0];   // [BATCH, SEQ] int32
    const float* w   = (const float*)d_in[1];   // [N_VOCAB, D_MODEL]
    const float* b   = (const float*)d_in[2];   // [SHARDS, D_MODEL]
    const float* pos = (const float*)d_in[3];   // [SHARDS, SEQ, D_PER_SHARD]
    float* out       = (float*)d_out;           // [BATCH, SEQ, D_MODEL]
    float* bias_sum  = (float*)d_ws;            // 4096 floats of scratch

    bias_sum_kernel<<<D_MODEL / 256, 256, 0, stream>>>(b, bias_sum);
    embed_add_kernel<<<SEQ, 512, 0, stream>>>(x, w, pos, bias_sum, out);
}